// SpatioTemporalModel_28630251995782
// MI455X (gfx1250) — compile-verified
//
#include <hip/hip_runtime.h>
#include <math.h>

// ============================================================================
// DiffPool GNN forward for MI455X (gfx1250), fp32 via V_WMMA_F32_16X16X4_F32.
// Working set (134MB adj + activations) is L2-resident (192MB) => compute
// bound on the matrix pipe. Each wave register-tiles 2x2 WMMA tiles (32x32)
// so A/B fragments are reused; A fragments load as 8B-aligned v2f (b64).
// Kernel is templated on transA/fused-root/bias/relu to kill branchy SALU.
// ============================================================================

typedef __attribute__((ext_vector_type(2))) float v2f;
typedef __attribute__((ext_vector_type(8))) float v8f;

#define BATCH 32

#define WMMA_F32(A_, B_, C_) \
  __builtin_amdgcn_wmma_f32_16x16x4_f32(false, (A_), false, (B_), (short)0, (C_), false, false)

// ---------------------------------------------------------------------------
// Generic batched GEMM:  C[b] = act( A[b](^T) @ Bm[b?]  (+ A2[b] @ B2) + bias )
// Block = 128 thr = 4 waves (2x2); each wave computes 32x32 (2x2 WMMA tiles).
// Requires M,N multiples of 64 (holds for every call), K,K2 multiples of 4,
// all leading dims even (holds: 64/192/256/384/1024) -> A v2f loads 8B-aligned.
// ---------------------------------------------------------------------------
template <bool TRANSA, bool HASA2, bool HASBIAS, bool RELU>
__global__ __launch_bounds__(128) void gemm_wmma_f32_kernel(
    const float* __restrict__ A, int lda, long sA,
    const float* __restrict__ Bm, int ldb, long sB,
    const float* __restrict__ A2, int lda2, long sA2,
    const float* __restrict__ B2, int ldb2,
    const float* __restrict__ bias,
    float* __restrict__ C, int ldc, long sC,
    int K, int K2)
{
  const int b    = blockIdx.z;
  const int wave = threadIdx.x >> 5;
  const int lane = threadIdx.x & 31;
  const int lm   = lane & 15;
  const int ko   = (lane >> 4) << 1;                 // 0 | 2 : K sub-offset
  const int m0   = blockIdx.y * 64 + (wave >> 1) * 32;
  const int n0   = blockIdx.x * 64 + (wave & 1) * 32;

  const float* Ab = A  + (size_t)b * sA;
  const float* Bb = Bm + (size_t)b * sB;

  v8f acc00 = {0.f,0.f,0.f,0.f,0.f,0.f,0.f,0.f};
  v8f acc01 = acc00, acc10 = acc00, acc11 = acc00;

  const float* bp0 = Bb + (size_t)ko * ldb + n0 + lm;
  const float* bp1 = bp0 + 16;

  if (!TRANSA) {
    const float* ar0 = Ab + (size_t)(m0 + lm) * lda + ko;        // 8B aligned
    const float* ar1 = ar0 + (size_t)16 * lda;
    #pragma unroll 2
    for (int k0 = 0; k0 < K; k0 += 4) {
      v2f a0 = *(const v2f*)(ar0 + k0);
      v2f a1 = *(const v2f*)(ar1 + k0);
      v2f b0, b1;
      b0.x = bp0[(size_t)k0 * ldb];  b0.y = bp0[(size_t)(k0 + 1) * ldb];
      b1.x = bp1[(size_t)k0 * ldb];  b1.y = bp1[(size_t)(k0 + 1) * ldb];
      acc00 = WMMA_F32(a0, b0, acc00);
      acc01 = WMMA_F32(a0, b1, acc01);
      acc10 = WMMA_F32(a1, b0, acc10);
      acc11 = WMMA_F32(a1, b1, acc11);
    }
  } else {
    // A element (m,k) = stored[k*lda + m]  (S^T @ X products)
    const float* ap0 = Ab + (size_t)ko * lda + m0 + lm;
    const float* ap1 = ap0 + 16;
    #pragma unroll 2
    for (int k0 = 0; k0 < K; k0 += 4) {
      v2f a0, a1, b0, b1;
      a0.x = ap0[(size_t)k0 * lda];  a0.y = ap0[(size_t)(k0 + 1) * lda];
      a1.x = ap1[(size_t)k0 * lda];  a1.y = ap1[(size_t)(k0 + 1) * lda];
      b0.x = bp0[(size_t)k0 * ldb];  b0.y = bp0[(size_t)(k0 + 1) * ldb];
      b1.x = bp1[(size_t)k0 * ldb];  b1.y = bp1[(size_t)(k0 + 1) * ldb];
      acc00 = WMMA_F32(a0, b0, acc00);
      acc01 = WMMA_F32(a0, b1, acc01);
      acc10 = WMMA_F32(a1, b0, acc10);
      acc11 = WMMA_F32(a1, b1, acc11);
    }
  }

  if (HASA2) {  // fused lin_root term: X @ Wroot
    const float* ar0 = A2 + (size_t)b * sA2 + (size_t)(m0 + lm) * lda2 + ko;
    const float* ar1 = ar0 + (size_t)16 * lda2;
    const float* cp0 = B2 + (size_t)ko * ldb2 + n0 + lm;
    const float* cp1 = cp0 + 16;
    #pragma unroll 2
    for (int k0 = 0; k0 < K2; k0 += 4) {
      v2f a0 = *(const v2f*)(ar0 + k0);
      v2f a1 = *(const v2f*)(ar1 + k0);
      v2f b0, b1;
      b0.x = cp0[(size_t)k0 * ldb2];  b0.y = cp0[(size_t)(k0 + 1) * ldb2];
      b1.x = cp1[(size_t)k0 * ldb2];  b1.y = cp1[(size_t)(k0 + 1) * ldb2];
      acc00 = WMMA_F32(a0, b0, acc00);
      acc01 = WMMA_F32(a0, b1, acc01);
      acc10 = WMMA_F32(a1, b0, acc10);
      acc11 = WMMA_F32(a1, b1, acc11);
    }
  }

  // ---- store 4 tiles: C/D layout VGPR g -> rows m0+g / m0+8+g ----
  float* Cb = C + (size_t)b * sC;
  const int rsub = (lane >> 4) << 3;  // +8 for upper half-wave
  v8f accs[2][2] = {{acc00, acc01}, {acc10, acc11}};
  #pragma unroll
  for (int mt = 0; mt < 2; ++mt) {
    #pragma unroll
    for (int nt = 0; nt < 2; ++nt) {
      const int col = n0 + nt * 16 + lm;
      const float bsv = HASBIAS ? bias[col] : 0.f;
      float* Cp = Cb + (size_t)(m0 + mt * 16 + rsub) * ldc + col;
      #pragma unroll
      for (int g = 0; g < 8; ++g) {
        float v = accs[mt][nt][g] + bsv;
        if (RELU) v = fmaxf(v, 0.f);
        Cp[(size_t)g * ldc] = v;
      }
    }
  }
}

// ---------------------------------------------------------------------------
// Link loss: accum += sum over tile of (adj[b,i,j] - (S @ S^T)[b,i,j])^2
// Both fragments come from contiguous S rows -> all loads are v2f (b64).
// 2x2 tiles per wave, 64x64 per block; wave reduction + one atomic per wave.
// ---------------------------------------------------------------------------
__global__ __launch_bounds__(128) void sst_link_loss_kernel(
    const float* __restrict__ S, int lds, long sS, int Kdim,
    const float* __restrict__ adj, int n, float* __restrict__ accum)
{
  const int b    = blockIdx.z;
  const int wave = threadIdx.x >> 5;
  const int lane = threadIdx.x & 31;
  const int lm   = lane & 15;
  const int ko   = (lane >> 4) << 1;
  const int m0   = blockIdx.y * 64 + (wave >> 1) * 32;
  const int n0   = blockIdx.x * 64 + (wave & 1) * 32;

  const float* Sb  = S + (size_t)b * sS;
  const float* ar0 = Sb + (size_t)(m0 + lm) * lds + ko;
  const float* ar1 = ar0 + (size_t)16 * lds;
  const float* br0 = Sb + (size_t)(n0 + lm) * lds + ko;   // B = S^T
  const float* br1 = br0 + (size_t)16 * lds;

  v8f acc00 = {0.f,0.f,0.f,0.f,0.f,0.f,0.f,0.f};
  v8f acc01 = acc00, acc10 = acc00, acc11 = acc00;
  #pragma unroll 2
  for (int k0 = 0; k0 < Kdim; k0 += 4) {
    v2f a0 = *(const v2f*)(ar0 + k0);
    v2f a1 = *(const v2f*)(ar1 + k0);
    v2f b0 = *(const v2f*)(br0 + k0);
    v2f b1 = *(const v2f*)(br1 + k0);
    acc00 = WMMA_F32(a0, b0, acc00);
    acc01 = WMMA_F32(a0, b1, acc01);
    acc10 = WMMA_F32(a1, b0, acc10);
    acc11 = WMMA_F32(a1, b1, acc11);
  }

  const float* adjb = adj + (size_t)b * n * n;
  const int rsub = (lane >> 4) << 3;
  v8f accs[2][2] = {{acc00, acc01}, {acc10, acc11}};
  float ss = 0.f;
  #pragma unroll
  for (int mt = 0; mt < 2; ++mt) {
    #pragma unroll
    for (int nt = 0; nt < 2; ++nt) {
      const int col = n0 + nt * 16 + lm;
      const float* ap = adjb + (size_t)(m0 + mt * 16 + rsub) * n + col;
      #pragma unroll
      for (int g = 0; g < 8; ++g) {
        float d = ap[(size_t)g * n] - accs[mt][nt][g];
        ss += d * d;
      }
    }
  }
  #pragma unroll
  for (int off = 16; off; off >>= 1) ss += __shfl_xor(ss, off, 32);
  if (lane == 0) atomicAdd(accum, ss);
}

// ---------------------------------------------------------------------------
// Row softmax (in place) + entropy accumulation. One wave per row.
// ---------------------------------------------------------------------------
__global__ __launch_bounds__(256) void softmax_ent_kernel(
    float* __restrict__ S, int L, int nrows, float* __restrict__ entAccum)
{
  const int row  = (int)((blockIdx.x * (unsigned)blockDim.x + threadIdx.x) >> 5);
  const int lane = threadIdx.x & 31;
  if (row >= nrows) return;
  float* r = S + (size_t)row * L;

  float mx = -INFINITY;
  for (int i = lane; i < L; i += 32) mx = fmaxf(mx, r[i]);
  #pragma unroll
  for (int off = 16; off; off >>= 1) mx = fmaxf(mx, __shfl_xor(mx, off, 32));

  float sum = 0.f;
  for (int i = lane; i < L; i += 32) sum += expf(r[i] - mx);
  #pragma unroll
  for (int off = 16; off; off >>= 1) sum += __shfl_xor(sum, off, 32);
  const float inv = 1.f / sum;

  float ent = 0.f;
  for (int i = lane; i < L; i += 32) {
    float s = expf(r[i] - mx) * inv;
    r[i] = s;
    ent -= s * logf(s + 1e-15f);
  }
  #pragma unroll
  for (int off = 16; off; off >>= 1) ent += __shfl_xor(ent, off, 32);
  if (lane == 0) atomicAdd(entAccum, ent);
}

// out[b,j] = sum_i X[b,i,j], X:[B,64,192] -> out:[B,192]
__global__ void readout_kernel(const float* __restrict__ X, float* __restrict__ out)
{
  int idx = blockIdx.x * blockDim.x + threadIdx.x;
  if (idx >= BATCH * 192) return;
  int b = idx / 192, j = idx % 192;
  const float* p = X + (size_t)b * 64 * 192 + j;
  float s = 0.f;
  #pragma unroll
  for (int i = 0; i < 64; ++i) s += p[i * 192];
  out[idx] = s;
}

__global__ void zero4_kernel(float* p) { if (threadIdx.x < 4) p[threadIdx.x] = 0.f; }

__global__ void finalize_kernel(const float* __restrict__ scal, float* __restrict__ out)
{
  if (threadIdx.x == 0) {
    out[6144] = sqrtf(scal[0]) / 33554432.f    // l1: adj.size = 32*1024*1024
              + sqrtf(scal[1]) / 2097152.f;    // l2: adj1.size = 32*256*256
    out[6145] = scal[2] / (32.f * 1024.f)      // e1: mean over B*1024 rows
              + scal[3] / (32.f * 256.f);      // e2: mean over B*256 rows
  }
}

// ---------------------------------------------------------------------------
// Host-side dispatch. Only 4 flag combinations are used.
// ---------------------------------------------------------------------------
static inline void launch_gemm(hipStream_t st,
    const float* A, int lda, long sA, int transA,
    const float* Bm, int ldb, long sB,
    const float* A2, int lda2, long sA2,
    const float* B2, int ldb2,
    const float* bias,
    float* C, int ldc, long sC,
    int M, int N, int K, int K2, int relu)
{
  dim3 grid(N / 64, M / 64, BATCH);
  if (transA) {
    gemm_wmma_f32_kernel<true, false, false, false><<<grid, 128, 0, st>>>(
        A, lda, sA, Bm, ldb, sB, A2, lda2, sA2, B2, ldb2, bias, C, ldc, sC, K, K2);
  } else if (A2 != nullptr) {
    gemm_wmma_f32_kernel<false, true, true, true><<<grid, 128, 0, st>>>(
        A, lda, sA, Bm, ldb, sB, A2, lda2, sA2, B2, ldb2, bias, C, ldc, sC, K, K2);
  } else if (bias != nullptr) {
    gemm_wmma_f32_kernel<false, false, true, true><<<grid, 128, 0, st>>>(
        A, lda, sA, Bm, ldb, sB, A2, lda2, sA2, B2, ldb2, bias, C, ldc, sC, K, K2);
  } else {
    gemm_wmma_f32_kernel<false, false, false, false><<<grid, 128, 0, st>>>(
        A, lda, sA, Bm, ldb, sB, A2, lda2, sA2, B2, ldb2, bias, C, ldc, sC, K, K2);
  }
}

extern "C" void kernel_launch(void* const* d_in, const int* in_sizes, int n_in,
                              void* d_out, int out_size, void* d_ws, size_t ws_size,
                              hipStream_t stream)
{
  const float* x   = (const float*)d_in[0];   // [32,1024,64]
  const float* adj = (const float*)d_in[1];   // [32,1024,1024]
  // params flattened in jax tree order (dict keys sorted):
  // gnn1_embed[0..8] gnn1_pool[9..19] gnn2_embed[20..28] gnn2_pool[29..39] gnn3_embed[40..48]
  // per conv: rel.b, rel.w, root.w ; per lin: b, w
  const float* const* P = (const float* const*)(d_in + 2);

  // ---- workspace layout (floats), heavy aliasing; total ~157 MB ----
  float* W     = (float*)d_ws;
  float* AX    = W;                  // 8,388,608  [B,1024,<=256] scratch product
  float* SCAT  = W + 8388608;        // 12,582,912 [B,1024,384] pool concat (lvl1)
  float* XE    = W + 20971520;       // 6,291,456  [B,1024,192] embed concat (lvl1)
  float* S     = W + 27262976;       // 8,388,608  [B,1024,256] assignment s
  float* X1    = W + 35651584;       // 1,572,864  [B,256,192]
  float* ADJ1  = W + 37224448;       // 2,097,152  [B,256,256]
  float* SCAL  = W + 39321600;       // 4: ss1, ss2, entsum1, entsum2
  // level 2/3 reuse the (freed) SCAT region:
  float* SCAT2 = SCAT;               // [B,256,192]
  float* XE2   = SCAT + 1572864;     // [B,256,192]
  float* S2    = SCAT + 3145728;     // [B,256,64]
  float* X2    = SCAT + 3670016;     // [B,64,192]
  float* ADJ2  = SCAT + 4063232;     // [B,64,64]
  float* SCAT3 = SCAT + 4194304;     // [B,64,192]
  if (ws_size < (size_t)(39321604) * sizeof(float)) return;  // need ~157MB

  zero4_kernel<<<1, 32, 0, stream>>>(SCAL);

  // ===================== Level 1 (n=1024) =====================
  // shared c1 aggregate: AX = adj @ x  [B,1024,64]
  launch_gemm(stream, adj,1024,1048576,0, x,64,65536, nullptr,0,0, nullptr,0, nullptr,
              AX,64,65536, 1024,64,1024,0,0);
  // embed c1 -> XE[:, :64]
  launch_gemm(stream, AX,64,65536,0, P[1],64,0, x,64,65536, P[2],64, P[0],
              XE,192,196608, 1024,64,64,64,1);
  // pool c1 -> SCAT[:, :64]
  launch_gemm(stream, AX,64,65536,0, P[10],64,0, x,64,65536, P[11],64, P[9],
              SCAT,384,393216, 1024,64,64,64,1);
  // pool c2
  launch_gemm(stream, adj,1024,1048576,0, SCAT,384,393216, nullptr,0,0, nullptr,0, nullptr,
              AX,64,65536, 1024,64,1024,0,0);
  launch_gemm(stream, AX,64,65536,0, P[13],64,0, SCAT,384,393216, P[14],64, P[12],
              SCAT+64,384,393216, 1024,64,64,64,1);
  // pool c3 (dout=256)
  launch_gemm(stream, adj,1024,1048576,0, SCAT+64,384,393216, nullptr,0,0, nullptr,0, nullptr,
              AX,64,65536, 1024,64,1024,0,0);
  launch_gemm(stream, AX,64,65536,0, P[16],256,0, SCAT+64,384,393216, P[17],256, P[15],
              SCAT+128,384,393216, 1024,256,64,64,1);
  // embed c2
  launch_gemm(stream, adj,1024,1048576,0, XE,192,196608, nullptr,0,0, nullptr,0, nullptr,
              AX,64,65536, 1024,64,1024,0,0);
  launch_gemm(stream, AX,64,65536,0, P[4],64,0, XE,192,196608, P[5],64, P[3],
              XE+64,192,196608, 1024,64,64,64,1);
  // embed c3
  launch_gemm(stream, adj,1024,1048576,0, XE+64,192,196608, nullptr,0,0, nullptr,0, nullptr,
              AX,64,65536, 1024,64,1024,0,0);
  launch_gemm(stream, AX,64,65536,0, P[7],64,0, XE+64,192,196608, P[8],64, P[6],
              XE+128,192,196608, 1024,64,64,64,1);
  // S = relu(SCAT @ linW + linb)  [B,1024,256], then softmax + ent
  launch_gemm(stream, SCAT,384,393216,0, P[19],256,0, nullptr,0,0, nullptr,0, P[18],
              S,256,262144, 1024,256,384,0,1);
  softmax_ent_kernel<<<dim3(32*1024/8), 256, 0, stream>>>(S, 256, 32*1024, SCAL+2);
  // X1 = S^T @ XE  [B,256,192]
  launch_gemm(stream, S,256,262144,1, XE,192,196608, nullptr,0,0, nullptr,0, nullptr,
              X1,192,49152, 256,192,1024,0,0);
  // AX = adj @ S   [B,1024,256]
  launch_gemm(stream, adj,1024,1048576,0, S,256,262144, nullptr,0,0, nullptr,0, nullptr,
              AX,256,262144, 1024,256,1024,0,0);
  // ADJ1 = S^T @ (adj @ S)  [B,256,256]
  launch_gemm(stream, S,256,262144,1, AX,256,262144, nullptr,0,0, nullptr,0, nullptr,
              ADJ1,256,65536, 256,256,1024,0,0);
  // SS1 = ||adj - S S^T||^2
  sst_link_loss_kernel<<<dim3(16,16,BATCH), 128, 0, stream>>>(S,256,262144, 256, adj, 1024, SCAL);

  // ===================== Level 2 (n=256) =====================
  launch_gemm(stream, ADJ1,256,65536,0, X1,192,49152, nullptr,0,0, nullptr,0, nullptr,
              AX,192,49152, 256,192,256,0,0);
  launch_gemm(stream, AX,192,49152,0, P[21],64,0, X1,192,49152, P[22],64, P[20],
              XE2,192,49152, 256,64,192,192,1);
  launch_gemm(stream, AX,192,49152,0, P[30],64,0, X1,192,49152, P[31],64, P[29],
              SCAT2,192,49152, 256,64,192,192,1);
  // pool c2, c3
  launch_gemm(stream, ADJ1,256,65536,0, SCAT2,192,49152, nullptr,0,0, nullptr,0, nullptr,
              AX,64,16384, 256,64,256,0,0);
  launch_gemm(stream, AX,64,16384,0, P[33],64,0, SCAT2,192,49152, P[34],64, P[32],
              SCAT2+64,192,49152, 256,64,64,64,1);
  launch_gemm(stream, ADJ1,256,65536,0, SCAT2+64,192,49152, nullptr,0,0, nullptr,0, nullptr,
              AX,64,16384, 256,64,256,0,0);
  launch_gemm(stream, AX,64,16384,0, P[36],64,0, SCAT2+64,192,49152, P[37],64, P[35],
              SCAT2+128,192,49152, 256,64,64,64,1);
  // embed c2, c3
  launch_gemm(stream, ADJ1,256,65536,0, XE2,192,49152, nullptr,0,0, nullptr,0, nullptr,
              AX,64,16384, 256,64,256,0,0);
  launch_gemm(stream, AX,64,16384,0, P[24],64,0, XE2,192,49152, P[25],64, P[23],
              XE2+64,192,49152, 256,64,64,64,1);
  launch_gemm(stream, ADJ1,256,65536,0, XE2+64,192,49152, nullptr,0,0, nullptr,0, nullptr,
              AX,64,16384, 256,64,256,0,0);
  launch_gemm(stream, AX,64,16384,0, P[27],64,0, XE2+64,192,49152, P[28],64, P[26],
              XE2+128,192,49152, 256,64,64,64,1);
  // S2 = relu(SCAT2 @ linW + linb) [B,256,64]; softmax + ent
  launch_gemm(stream, SCAT2,192,49152,0, P[39],64,0, nullptr,0,0, nullptr,0, P[38],
              S2,64,16384, 256,64,192,0,1);
  softmax_ent_kernel<<<dim3(32*256/8), 256, 0, stream>>>(S2, 64, 32*256, SCAL+3);
  // X2 = S2^T @ XE2 [B,64,192]
  launch_gemm(stream, S2,64,16384,1, XE2,192,49152, nullptr,0,0, nullptr,0, nullptr,
              X2,192,12288, 64,192,256,0,0);
  // ADJ2 = S2^T @ (ADJ1 @ S2) [B,64,64]
  launch_gemm(stream, ADJ1,256,65536,0, S2,64,16384, nullptr,0,0, nullptr,0, nullptr,
              AX,64,16384, 256,64,256,0,0);
  launch_gemm(stream, S2,64,16384,1, AX,64,16384, nullptr,0,0, nullptr,0, nullptr,
              ADJ2,64,4096, 64,64,256,0,0);
  sst_link_loss_kernel<<<dim3(4,4,BATCH), 128, 0, stream>>>(S2,64,16384, 64, ADJ1, 256, SCAL+1);

  // ===================== Level 3 (n=64) =====================
  launch_gemm(stream, ADJ2,64,4096,0, X2,192,12288, nullptr,0,0, nullptr,0, nullptr,
              AX,192,12288, 64,192,64,0,0);
  launch_gemm(stream, AX,192,12288,0, P[41],64,0, X2,192,12288, P[42],64, P[40],
              SCAT3,192,12288, 64,64,192,192,1);
  launch_gemm(stream, ADJ2,64,4096,0, SCAT3,192,12288, nullptr,0,0, nullptr,0, nullptr,
              AX,64,4096, 64,64,64,0,0);
  launch_gemm(stream, AX,64,4096,0, P[44],64,0, SCAT3,192,12288, P[45],64, P[43],
              SCAT3+64,192,12288, 64,64,64,64,1);
  launch_gemm(stream, ADJ2,64,4096,0, SCAT3+64,192,12288, nullptr,0,0, nullptr,0, nullptr,
              AX,64,4096, 64,64,64,0,0);
  launch_gemm(stream, AX,64,4096,0, P[47],64,0, SCAT3+64,192,12288, P[48],64, P[46],
              SCAT3+128,192,12288, 64,64,64,64,1);

  // readout + losses
  readout_kernel<<<(BATCH*192 + 255)/256, 256, 0, stream>>>(SCAT3, (float*)d_out);
  finalize_kernel<<<1, 32, 0, stream>>>(SCAL, (float*)d_out);
}